// SSL_9552007266831
// MI455X (gfx1250) — compile-verified
//
#include <hip/hip_runtime.h>

// ---------------------------------------------------------------------------
// SS1 "universal-hash compressed linear" for MI455X (gfx1250, wave32).
//   y[64,8192] = x[64,8192] @ gather(weight[8192,2048], h)ᵀ
// Rewritten as:  xc[m,c] = Σ_{j:h(j)=c} x[m,j]   (scatter-add, LDS)
//                y       = xc @ weightᵀ          (V_WMMA_F32_16X16X4_F32)
// ---------------------------------------------------------------------------

typedef float v2f __attribute__((ext_vector_type(2)));
typedef float v8f __attribute__((ext_vector_type(8)));

#define K_FULL   8192
#define K_COMP   2048
#define M_ROWS   64
#define N_COLS   8192
#define P_MERS   0x7FFFFFFFu

__device__ __forceinline__ unsigned hash_col(unsigned long long r0,
                                             unsigned long long r1,
                                             unsigned j) {
    unsigned long long v = r0 * (unsigned long long)j + r1;   // < 2^34
    // v mod (2^31-1):  a*2^31 + b == a + b (mod P), fold twice + cond-sub
    v = (v >> 31) + (v & P_MERS);
    v = (v >> 31) + (v & P_MERS);
    if (v >= P_MERS) v -= P_MERS;
    return (unsigned)v & (K_COMP - 1);                        // mod 2048
}

// ---- Kernel 1: compress x along K via the hash map (one block per row m) ---
__global__ __launch_bounds__(256)
void ssl_compress_x(const float* __restrict__ x,
                    const long long* __restrict__ rn,   // int64 per reference
                    float* __restrict__ xc) {
    __shared__ float acc[K_COMP];                       // 8 KB of 320 KB LDS
    const int m   = blockIdx.x;
    const int tid = threadIdx.x;

    for (int c = tid; c < K_COMP; c += 256) acc[c] = 0.0f;
    __syncthreads();

    const unsigned long long r0 = (unsigned long long)rn[0];
    const unsigned long long r1 = (unsigned long long)rn[1];
    const float* xrow = x + (size_t)m * K_FULL;

    #pragma unroll 4
    for (int j = tid; j < K_FULL; j += 256) {
        const unsigned c = hash_col(r0, r1, (unsigned)j);
        atomicAdd(&acc[c], xrow[j]);                    // ds_add_f32
    }
    __syncthreads();

    float* xcrow = xc + (size_t)m * K_COMP;
    for (int c = tid; c < K_COMP; c += 256) xcrow[c] = acc[c];
}

// ---- Kernel 2: y = xc @ weightᵀ with V_WMMA_F32_16X16X4_F32 ----------------
// Wave -> (ntile in 0..511, mhalf in 0..1). Each wave: two 16x16 C tiles,
// K=2048 in steps of 4. Lane layout (ISA 7.12.2, 32-bit A 16x4 / B 4x16):
//   row  = lane % 16, k-pair offset = 2*(lane/16)  -> one float2 per operand.
__global__ __launch_bounds__(256)
void ssl_gemm_wmma(const float* __restrict__ xc,
                   const float* __restrict__ w,
                   float* __restrict__ y) {
    const int lane  = threadIdx.x & 31;
    const int wave  = threadIdx.x >> 5;
    const int gw    = blockIdx.x * 8 + wave;   // 0..1023
    const int ntile = gw >> 1;                 // 0..511
    const int mhalf = gw & 1;                  // 0..1
    const int row   = lane & 15;
    const int koff  = (lane >> 4) << 1;        // 0 or 2

    const float* brow  = w  + (size_t)(ntile * 16 + row) * K_COMP + koff;
    const float* arow0 = xc + (size_t)(mhalf * 32 + row) * K_COMP + koff;
    const float* arow1 = arow0 + (size_t)16 * K_COMP;

    v8f c0 = {};
    v8f c1 = {};

    #pragma unroll 4
    for (int k = 0; k < K_COMP; k += 4) {
        const v2f b  = *(const v2f*)(brow  + k);   // global_load_b64
        const v2f a0 = *(const v2f*)(arow0 + k);
        const v2f a1 = *(const v2f*)(arow1 + k);
        c0 = __builtin_amdgcn_wmma_f32_16x16x4_f32(false, a0, false, b,
                                                   (short)0, c0, false, false);
        c1 = __builtin_amdgcn_wmma_f32_16x16x4_f32(false, a1, false, b,
                                                   (short)0, c1, false, false);
    }

    // C/D layout: VGPR v holds M = v + 8*(lane>=16), N = lane%16.
    const int n      = ntile * 16 + row;
    const int mbase  = mhalf * 32 + ((lane >> 4) << 3);
    #pragma unroll
    for (int v = 0; v < 8; ++v) {
        y[(size_t)(mbase +      v) * N_COLS + n] = c0[v];
        y[(size_t)(mbase + 16 + v) * N_COLS + n] = c1[v];
    }
}

// ---------------------------------------------------------------------------
extern "C" void kernel_launch(void* const* d_in, const int* in_sizes, int n_in,
                              void* d_out, int out_size, void* d_ws, size_t ws_size,
                              hipStream_t stream) {
    const float*     x  = (const float*)d_in[0];          // [64, 8192] f32
    const float*     w  = (const float*)d_in[1];          // [8192, 2048] f32
    const long long* rn = (const long long*)d_in[2];      // [4] int64
    float*           y  = (float*)d_out;                  // [64, 8192] f32
    float*           xc = (float*)d_ws;                   // [64, 2048] f32 scratch

    ssl_compress_x<<<M_ROWS, 256, 0, stream>>>(x, rn, xc);
    ssl_gemm_wmma<<<128, 256, 0, stream>>>(xc, w, y);
}